// WGCN_45827301048726
// MI455X (gfx1250) — compile-verified
//
#include <hip/hip_runtime.h>

#define NN 50000
#define NE 800000
#define DIM 128
#define LN_EPS 1e-5f

typedef __attribute__((ext_vector_type(2))) float v2f;
typedef __attribute__((ext_vector_type(8))) float v8f;

// ---------------- utility: zero fill ----------------
__global__ void wgcn_fill_zero(float* __restrict__ p, int n) {
  int i = blockIdx.x * blockDim.x + threadIdx.x;
  if (i < n) p[i] = 0.0f;
}

// ---------------- edge stats: weighted + structural degrees ----------------
__global__ void wgcn_edge_stats(const float* __restrict__ ew,
                                const int* __restrict__ src,
                                const int* __restrict__ dst,
                                float* __restrict__ out_w, float* __restrict__ in_w,
                                float* __restrict__ deg_out, float* __restrict__ deg_in,
                                int E) {
  int e = blockIdx.x * blockDim.x + threadIdx.x;
  if (e >= E) return;
  float w = ew[e];
  int s = src[e];
  int d = dst[e];
  atomicAdd(&out_w[s], w);
  atomicAdd(&in_w[d], w);
  atomicAdd(&deg_out[s], 1.0f);
  atomicAdd(&deg_in[d], 1.0f);
}

// ---------------- per-node rsqrt norms (in place) ----------------
__global__ void wgcn_node_norms(float* __restrict__ out_w, float* __restrict__ in_w,
                                float* __restrict__ deg_out, float* __restrict__ deg_in,
                                int n) {
  int i = blockIdx.x * blockDim.x + threadIdx.x;
  if (i >= n) return;
  out_w[i]   = rsqrtf(out_w[i]);                 // unused for isolated nodes
  in_w[i]    = rsqrtf(in_w[i]);
  deg_out[i] = rsqrtf(fmaxf(deg_out[i], 1.0f));  // dn_out
  deg_in[i]  = rsqrtf(fmaxf(deg_in[i], 1.0f));   // dn_in
}

// ---------------- layer-invariant per-edge coefficient ----------------
// c[e] = w[e] * rsqrt(out_w[src]) * rsqrt(in_w[dst]) * dn_out[src]
__global__ void wgcn_edge_coef(const float* __restrict__ ew,
                               const int* __restrict__ src,
                               const int* __restrict__ dst,
                               const float* __restrict__ rs_out,
                               const float* __restrict__ rs_in,
                               const float* __restrict__ dn_out,
                               float* __restrict__ coef, int E) {
  int e = blockIdx.x * blockDim.x + threadIdx.x;
  if (e >= E) return;
  int s = src[e];
  coef[e] = ew[e] * rs_out[s] * rs_in[dst[e]] * dn_out[s];
}

// ---------------- SpMM: one wave per edge, float4 per lane ----------------
__global__ void wgcn_spmm(const float* __restrict__ x, const float* __restrict__ coef,
                          const int* __restrict__ src, const int* __restrict__ dst,
                          float* __restrict__ agg, int E) {
  int gid  = blockIdx.x * blockDim.x + threadIdx.x;
  int e    = gid >> 5;
  int lane = gid & 31;
  if (e >= E) return;
  int s = src[e];
  int d = dst[e];
  float w = coef[e];
  const float4* xs = reinterpret_cast<const float4*>(x + (size_t)s * DIM);
  float4 v = xs[lane];                      // cols 4*lane .. 4*lane+3 (coalesced)
  float* o = agg + (size_t)d * DIM + lane * 4;
  atomicAdd(o + 0, v.x * w);
  atomicAdd(o + 1, v.y * w);
  atomicAdd(o + 2, v.z * w);
  atomicAdd(o + 3, v.w * w);
}

// ---------------- fused (agg*dn_in) @ W + b -> relu -> layernorm ----------------
// Block: 16 rows x HOUT cols; one wave per 16x16 output tile; fp32 WMMA (K=4).
template <int HOUT>
__global__ __launch_bounds__((HOUT / 16) * 32)
void wgcn_gemm_ln(const float* __restrict__ agg, const float* __restrict__ dn_in,
                  const float* __restrict__ W, const float* __restrict__ bias,
                  const float* __restrict__ lng, const float* __restrict__ lnb,
                  float* __restrict__ out) {
  constexpr int NWAVES   = HOUT / 16;
  constexpr int NTHREADS = NWAVES * 32;
  constexpr int APITCH   = DIM + 4;    // conflict-free fragment reads
  constexpr int OPITCH   = HOUT + 4;
  constexpr int STRIDE   = NTHREADS / 16;

  __shared__ float a_lds[16 * APITCH];
  __shared__ float o_lds[16 * OPITCH];
  __shared__ float red_s[NTHREADS];
  __shared__ float red_q[NTHREADS];
  __shared__ float mu_s[16];
  __shared__ float rs_s[16];

  const int tid  = threadIdx.x;
  const int row0 = blockIdx.x * 16;

  // Stage A tile (16 x 128), pre-scaled by dn_in[row]  (coalesced global reads)
  for (int e = tid; e < 16 * DIM; e += NTHREADS) {
    int r = e >> 7;
    int k = e & (DIM - 1);
    a_lds[r * APITCH + k] = agg[(size_t)(row0 + r) * DIM + k] * dn_in[row0 + r];
  }
  __syncthreads();

  const int wave = tid >> 5;
  const int lane = tid & 31;
  const int half = lane >> 4;   // 0: lanes 0-15, 1: lanes 16-31
  const int lm   = lane & 15;
  const int n0   = wave * 16;

  v8f acc = {0.f, 0.f, 0.f, 0.f, 0.f, 0.f, 0.f, 0.f};
#pragma unroll
  for (int kk = 0; kk < DIM; kk += 4) {
    const int k = kk + 2 * half;
    // A fragment: A[lm][k], A[lm][k+1]  (8B-aligned LDS read)
    v2f a = *reinterpret_cast<const v2f*>(&a_lds[lm * APITCH + k]);
    // B fragment: W[k][n0+lm], W[k+1][n0+lm]
    v2f b;
    b.x = W[(size_t)k * HOUT + n0 + lm];
    b.y = W[(size_t)(k + 1) * HOUT + n0 + lm];
    acc = __builtin_amdgcn_wmma_f32_16x16x4_f32(false, a, false, b,
                                                (short)0, acc, false, false);
  }

  // bias + relu into LDS tile:  D[v + 8*half][n0+lm] = acc[v]
  const int col = n0 + lm;
  const float bv = bias[col];
#pragma unroll
  for (int v = 0; v < 8; ++v) {
    int r = v + 8 * half;
    o_lds[r * OPITCH + col] = fmaxf(acc[v] + bv, 0.0f);
  }
  __syncthreads();

  // layernorm: per-row partial sums
  {
    const int r  = tid & 15;
    const int j0 = tid >> 4;
    float s = 0.f, q = 0.f;
    for (int j = j0; j < HOUT; j += STRIDE) {
      float v = o_lds[r * OPITCH + j];
      s += v;
      q += v * v;
    }
    red_s[tid] = s;
    red_q[tid] = q;
  }
  __syncthreads();
  if (tid < 16) {
    float s = 0.f, q = 0.f;
    for (int i = 0; i < STRIDE; ++i) {
      s += red_s[i * 16 + tid];
      q += red_q[i * 16 + tid];
    }
    float mu  = s / (float)HOUT;
    float var = q / (float)HOUT - mu * mu;
    mu_s[tid] = mu;
    rs_s[tid] = rsqrtf(var + LN_EPS);
  }
  __syncthreads();

  // normalize + affine, coalesced store
  for (int e = tid; e < 16 * HOUT; e += NTHREADS) {
    int r = e / HOUT;
    int j = e % HOUT;
    float v = o_lds[r * OPITCH + j];
    out[(size_t)(row0 + r) * HOUT + j] = (v - mu_s[r]) * rs_s[r] * lng[j] + lnb[j];
  }
}

extern "C" void kernel_launch(void* const* d_in, const int* in_sizes, int n_in,
                              void* d_out, int out_size, void* d_ws, size_t ws_size,
                              hipStream_t stream) {
  const float* feats = (const float*)d_in[0];
  const float* ew    = (const float*)d_in[1];
  const int*   src   = (const int*)d_in[2];
  const int*   dst   = (const int*)d_in[3];
  const float* W1 = (const float*)d_in[4];
  const float* b1 = (const float*)d_in[5];
  const float* W2 = (const float*)d_in[6];
  const float* b2 = (const float*)d_in[7];
  const float* W3 = (const float*)d_in[8];
  const float* b3 = (const float*)d_in[9];
  const float* ln1g = (const float*)d_in[10];
  const float* ln1b = (const float*)d_in[11];
  const float* ln2g = (const float*)d_in[12];
  const float* ln2b = (const float*)d_in[13];
  const float* ln3g = (const float*)d_in[14];
  const float* ln3b = (const float*)d_in[15];
  float* out = (float*)d_out;

  // workspace carve-up (floats)
  float* ws    = (float*)d_ws;
  float* out_w = ws;  ws += NN;            // -> rsqrt(out_w)
  float* in_w  = ws;  ws += NN;            // -> rsqrt(in_w)
  float* dnout = ws;  ws += NN;            // -> dn_out
  float* dnin  = ws;  ws += NN;            // -> dn_in
  float* coef  = ws;  ws += NE;            // layer-invariant edge coefficient
  float* agg   = ws;  ws += (size_t)NN * DIM;
  float* hbuf  = ws;  ws += (size_t)NN * DIM;  // h1 then h2

  const int AGG_ELEMS   = NN * DIM;            // 6.4M
  const int spmm_blocks = NE / 8;              // 8 edges (waves) per 256-thr block
  const int gemm_blocks = NN / 16;             // 3125 (exact)

  // --- graph norms (once) ---
  wgcn_fill_zero<<<(4 * NN + 255) / 256, 256, 0, stream>>>(out_w, 4 * NN);
  wgcn_edge_stats<<<(NE + 255) / 256, 256, 0, stream>>>(ew, src, dst, out_w, in_w,
                                                        dnout, dnin, NE);
  wgcn_node_norms<<<(NN + 255) / 256, 256, 0, stream>>>(out_w, in_w, dnout, dnin, NN);
  wgcn_edge_coef<<<(NE + 255) / 256, 256, 0, stream>>>(ew, src, dst, out_w, in_w,
                                                       dnout, coef, NE);

  // --- layer 1 ---
  wgcn_fill_zero<<<(AGG_ELEMS + 255) / 256, 256, 0, stream>>>(agg, AGG_ELEMS);
  wgcn_spmm<<<spmm_blocks, 256, 0, stream>>>(feats, coef, src, dst, agg, NE);
  wgcn_gemm_ln<128><<<gemm_blocks, 256, 0, stream>>>(agg, dnin, W1, b1, ln1g, ln1b, hbuf);

  // --- layer 2 ---
  wgcn_fill_zero<<<(AGG_ELEMS + 255) / 256, 256, 0, stream>>>(agg, AGG_ELEMS);
  wgcn_spmm<<<spmm_blocks, 256, 0, stream>>>(hbuf, coef, src, dst, agg, NE);
  wgcn_gemm_ln<128><<<gemm_blocks, 256, 0, stream>>>(agg, dnin, W2, b2, ln2g, ln2b, hbuf);

  // --- layer 3 (HOUT = 256, output straight to d_out) ---
  wgcn_fill_zero<<<(AGG_ELEMS + 255) / 256, 256, 0, stream>>>(agg, AGG_ELEMS);
  wgcn_spmm<<<spmm_blocks, 256, 0, stream>>>(hbuf, coef, src, dst, agg, NE);
  wgcn_gemm_ln<256><<<gemm_blocks, 512, 0, stream>>>(agg, dnin, W3, b3, ln3g, ln3b, out);
}